// PolynomialFlowRegularizer_38654705664138
// MI455X (gfx1250) — compile-verified
//
#include <hip/hip_runtime.h>

typedef __attribute__((ext_vector_type(2))) float v2f;
typedef __attribute__((ext_vector_type(8))) float v8f;

#define BATCH 64
#define CH 2
#define HDIM 512
#define WDIM 512
#define NPIX (HDIM * WDIM)          // 262144 elements per (b,c) row
#define BCROWS (BATCH * CH)         // 128 rows of the implicit GEMM
#define ROWG (BCROWS / 16)          // 8 groups of 16 bc-rows (WMMA M tile)
#define YSLICES HDIM                // one image row per wave -> 512 K-slices
#define NWAVES (ROWG * YSLICES)     // 4096 waves
#define TILE_F 288                  // floats per partial tile: 256 (D) + 32 (s2)
#define NCHUNK 16                   // 128 WMMA steps = 16 chunks of 8

// ---------------------------------------------------------------------------
// Kernel 1: bulk pass. D = flow_tile(16x4 f32) x basis_tile(4x16 f32) + D via
// V_WMMA_F32_16X16X4_F32, accumulating rhs[16 bc-rows][6 basis cols] per wave,
// plus per-lane sum of flow^2. The basis operand is generated by a 2nd-order
// recurrence (zero memory traffic for B). The chunk loop is fully unrolled and
// explicitly double-buffered so each wave keeps ~16 b64 loads (4 KB) in
// flight while the WMMAs consume the previous chunk; with 4096 waves that is
// ~16 MB of outstanding reads device-wide — enough to cover HBM latency at
// 23.3 TB/s.
//
// Lane mapping (ISA 7.12.2, wave32):
//   A 16x4 f32 : lane L, vgpr j -> A[M = L%16][K = j + 2*(L/16)]
//   B  4x16 f32: lane L, vgpr j -> B[K = j + 2*(L/16)][Ncol = L%16]
//   C/D 16x16  : lane L, vgpr j -> D[M = j + 8*(L/16)][Ncol = L%16]
// A lane's two A values and two B values refer to the SAME two n positions
// (n = y*512 + 4t + 2*half + j), which keeps the basis recurrence per-lane.
// ---------------------------------------------------------------------------
__global__ __launch_bounds__(256)
void poly_rhs_wmma(const float* __restrict__ flow, float* __restrict__ ws) {
    const int lane = threadIdx.x & 31;
    const int wid  = (int)((blockIdx.x * blockDim.x + threadIdx.x) >> 5);
    const int g    = wid >> 9;                  // bc row-group (YSLICES = 512)
    const int y    = wid & (YSLICES - 1);       // image row = K-slice
    const int half = lane >> 4;                 // 0: K={0,1}, 1: K={2,3}
    const int m    = lane & 15;                 // A row in group == B column

    const float DX    = 2.0f / (float)(WDIM - 1);
    const float DY    = 2.0f / (float)(HDIM - 1);
    const float STEPX = 4.0f * DX;              // x advance per WMMA step
    // only the x^2 column (col==3) has a non-constant first difference
    const float ddv = (m == 3) ? (2.0f * STEPX * STEPX) : 0.0f;
    const float yf  = -1.0f + (float)y * DY;

    // init basis recurrence for this image row (branchless: selects only)
    float v[2], dv[2];
#pragma unroll
    for (int j = 0; j < 2; ++j) {
        const float xf = -1.0f + (float)(2 * half + j) * DX;
        v[j]  = (m == 0) ? 1.0f
              : (m == 1) ? xf
              : (m == 2) ? yf
              : (m == 3) ? xf * xf
              : (m == 4) ? xf * yf
              : (m == 5) ? yf * yf
              : 0.0f;
        dv[j] = (m == 1) ? STEPX
              : (m == 3) ? STEPX * (2.0f * xf + STEPX)
              : (m == 4) ? STEPX * yf
              : 0.0f;
    }

    v8f acc = {};       // rhs partial tile (16 bc-rows x 16 cols, cols>=6 stay 0)
    float s2 = 0.0f;    // per-lane partial sum of flow^2

    const float* p = flow + (size_t)(16 * g + m) * (size_t)NPIX
                          + (size_t)y * WDIM + (size_t)(2 * half);

    // double-buffered, fully unrolled chunk pipeline
    float2 buf[2][8];
#pragma unroll
    for (int u = 0; u < 8; ++u)
        buf[0][u] = *(const float2*)(p + 4 * u);

#pragma unroll
    for (int c = 0; c < NCHUNK; ++c) {
        if (c + 1 < NCHUNK) {
#pragma unroll
            for (int u = 0; u < 8; ++u)
                buf[(c + 1) & 1][u] = *(const float2*)(p + 32 * (c + 1) + 4 * u);
        }
#pragma unroll
        for (int u = 0; u < 8; ++u) {
            const float2 a2 = buf[c & 1][u];
            v2f a; a[0] = a2.x; a[1] = a2.y;                 // A: flow tile
            v2f b; b[0] = v[0]; b[1] = v[1];                 // B: basis tile
            acc = __builtin_amdgcn_wmma_f32_16x16x4_f32(
                false, a, false, b, (short)0, acc, false, false);
            s2 = fmaf(a2.x, a2.x, s2);
            s2 = fmaf(a2.y, a2.y, s2);
            v[0] += dv[0]; dv[0] += ddv;                     // basis recurrence
            v[1] += dv[1]; dv[1] += ddv;
        }
    }

    float* tile = ws + (size_t)wid * TILE_F;
#pragma unroll
    for (int j = 0; j < 8; ++j) tile[lane * 8 + j] = acc[j];
    tile[256 + lane] = s2;
}

// ---------------------------------------------------------------------------
// Kernel 2: deterministic fixed-order reduction of partials + closed-form
// normal-equations solve. G is block-diagonal on a symmetric separable grid:
// odd moments vanish, leaving a 3x3 system over {1, x^2, y^2} and scalars for
// {x, y, xy}. err_bc = (sum flow^2 - rhs.coeff)/N; out = sum_bc err / B.
// ---------------------------------------------------------------------------
__global__ __launch_bounds__(128)
void poly_finalize(const float* __restrict__ ws, float* __restrict__ out) {
    const int bc = threadIdx.x;       // 0..127
    const int g  = bc >> 4;
    const int m  = bc & 15;

    float rhs[6] = {0.f, 0.f, 0.f, 0.f, 0.f, 0.f};
    float s2 = 0.0f;
    for (int s = 0; s < YSLICES; ++s) {
        const float* tile = ws + (size_t)(g * YSLICES + s) * TILE_F;
#pragma unroll
        for (int k = 0; k < 6; ++k)
            rhs[k] += tile[(k + 16 * (m >> 3)) * 8 + (m & 7)];
        s2 += tile[256 + m] + tile[256 + 16 + m];     // both lane halves of row m
    }

    // grid moments (identical for x and y: 512 symmetric points in [-1,1])
    double S2 = 0.0, S4 = 0.0;
    const double dxi = 2.0 / (double)(WDIM - 1);
    for (int i = 0; i < WDIM; ++i) {
        const double xf = -1.0 + (double)i * dxi;
        const double x2 = xf * xf;
        S2 += x2;
        S4 += x2 * x2;
    }
    const double g00 = (double)NPIX;        // <1,1>
    const double g03 = (double)HDIM * S2;   // <1,x^2> = <1,y^2> = <x,x> = <y,y>
    const double g33 = (double)HDIM * S4;   // <x^2,x^2> = <y^2,y^2>
    const double g35 = S2 * S2;             // <x^2,y^2> = <xy,xy>

    // Cramer solve of symmetric 3x3 for columns {0:1, 3:x^2, 5:y^2}
    const double r0 = rhs[0], r3 = rhs[3], r5 = rhs[5];
    const double A00 = g00, A01 = g03, A02 = g03, A11 = g33, A12 = g35, A22 = g33;
    const double det = A00 * (A11 * A22 - A12 * A12)
                     - A01 * (A01 * A22 - A12 * A02)
                     + A02 * (A01 * A12 - A11 * A02);
    const double c0 = ( r0 * (A11 * A22 - A12 * A12)
                      - A01 * (r3 * A22 - A12 * r5)
                      + A02 * (r3 * A12 - A11 * r5)) / det;
    const double c3 = ( A00 * (r3 * A22 - r5 * A12)
                      - r0 * (A01 * A22 - A12 * A02)
                      + A02 * (A01 * r5 - r3 * A02)) / det;
    const double c5 = ( A00 * (A11 * r5 - A12 * r3)
                      - A01 * (A01 * r5 - r3 * A02)
                      + r0 * (A01 * A12 - A11 * A02)) / det;

    const double energy = r0 * c0 + r3 * c3 + r5 * c5
                        + (double)rhs[1] * rhs[1] / g03     // x column
                        + (double)rhs[2] * rhs[2] / g03     // y column
                        + (double)rhs[4] * rhs[4] / g35;    // xy column
    const float err = (float)(((double)s2 - energy) / (double)NPIX);

    __shared__ float red[128];
    red[bc] = err;
    __syncthreads();
    for (int stride = 64; stride > 0; stride >>= 1) {
        if (bc < stride) red[bc] += red[bc + stride];
        __syncthreads();
    }
    if (bc == 0) out[0] = red[0] / (float)BATCH;
}

extern "C" void kernel_launch(void* const* d_in, const int* in_sizes, int n_in,
                              void* d_out, int out_size, void* d_ws, size_t ws_size,
                              hipStream_t stream) {
    const float* flow = (const float*)d_in[0];
    float* ws  = (float*)d_ws;     // needs NWAVES * TILE_F * 4 = 4.5 MiB
    float* out = (float*)d_out;

    // 4096 waves, 8 waves (256 threads) per block -> 512 blocks
    poly_rhs_wmma<<<(NWAVES * 32) / 256, 256, 0, stream>>>(flow, ws);
    poly_finalize<<<1, 128, 0, stream>>>(ws, out);
}